// Upsample_41506563948884
// MI455X (gfx1250) — compile-verified
//
#include <hip/hip_runtime.h>
#include <cstdint>

// ---- shapes fixed by the reference ----
#define B_DIM   8
#define C_DIM   16
#define H_IN    384
#define W_IN    384
#define H_OUT   768
#define W_OUT   768
#define NPLANES (B_DIM * C_DIM)            // 128
#define THREADS 256                        // 8 wave32 waves

// ---- tiling ----
#define TILE_ROWS_OUT 16                   // output rows per tile
#define TILE_ROWS_IN  10                   // input rows staged per tile (span<=9 + slack)
#define SPLIT         8                    // row-strips per plane
#define ROWS_PER_BLOCK (H_OUT / SPLIT)     // 96
#define TILES          (ROWS_PER_BLOCK / TILE_ROWS_OUT) // 6

typedef __attribute__((ext_vector_type(4))) unsigned int u32x4;
typedef __attribute__((ext_vector_type(8))) int          i32x8;
typedef __attribute__((ext_vector_type(4))) int          i32x4;

// Issue one TDM DMA: TILE_ROWS_IN x W_IN f32 tile starting at input row
// ihFirst of this plane -> LDS at lds_byte_off. Rows past the tensor edge
// read as zeros (their interpolation weight is exactly 0 in those cases).
__device__ __forceinline__ void tdm_issue_rows(const float* gsrc_plane,
                                               int ihFirst,
                                               uint32_t lds_byte_off) {
#if defined(__HIP_DEVICE_COMPILE__)
    const uint64_t gaddr =
        (uint64_t)(uintptr_t)(gsrc_plane + (size_t)ihFirst * W_IN);
    const uint32_t validRows = (uint32_t)(H_IN - ihFirst);  // OOB rows -> 0

    // D# group 0: count=1 (valid), lds_addr, global_addr[56:0], type=2
    u32x4 g0;
    g0.x = 1u;
    g0.y = lds_byte_off;
    g0.z = (uint32_t)gaddr;
    g0.w = (uint32_t)((gaddr >> 32) & 0x1FFFFFFull) | (2u << 30);

    // D# group 1: data_size=4B, tensor 384 x validRows, tile 384 x 10, stride 384
    i32x8 g1;
    g1[0] = (int)(2u << 16);                                  // data_size=2 -> 4B
    g1[1] = (int)(((unsigned)W_IN & 0xFFFFu) << 16);          // tensor_dim0[15:0]
    g1[2] = (int)((validRows & 0xFFFFu) << 16);               // dim0 hi=0 | tensor_dim1 lo
    g1[3] = (int)(((validRows >> 16) & 0xFFFFu)
                  | (((unsigned)W_IN & 0xFFFFu) << 16));      // dim1 hi | tile_dim0
    g1[4] = TILE_ROWS_IN;                                     // tile_dim1 | tile_dim2=0
    g1[5] = W_IN;                                             // tensor_dim0_stride lo32
    g1[6] = 0;
    g1[7] = 0;

    const i32x4 gz = {0, 0, 0, 0};                            // 2D: groups 2/3 unused
#if __clang_major__ >= 23
    const i32x8 gz8 = {0, 0, 0, 0, 0, 0, 0, 0};
    __builtin_amdgcn_tensor_load_to_lds(g0, g1, gz, gz, gz8, 0);
#else
    __builtin_amdgcn_tensor_load_to_lds(g0, g1, gz, gz, 0);
#endif
#else
    (void)gsrc_plane; (void)ihFirst; (void)lds_byte_off;
#endif
}

__global__ __launch_bounds__(THREADS)
void upsample2x_tdm_pipe(const float* __restrict__ x, float* __restrict__ out) {
    __shared__ float sbuf[2][TILE_ROWS_IN * W_IN];   // 2 x 15 KB double buffer

    const int plane = blockIdx.x;                    // 0..127
    const int rbase = blockIdx.y * ROWS_PER_BLOCK;   // first output row of strip
    const float s   = (float)(H_IN - 1) / (float)(H_OUT - 1);   // 383/767
    const float* psrc = x + (size_t)plane * (H_IN * W_IN);

    // Horizontal taps for this thread's 3 columns (row-invariant).
    int   iw0[3], iw1[3];
    float ww[3];
#pragma unroll
    for (int k = 0; k < 3; ++k) {
        const int   j  = (int)threadIdx.x + k * THREADS;
        const float cw = (float)j * s;
        int i0 = (int)floorf(cw);
        if (i0 > W_IN - 1) i0 = W_IN - 1;
        ww[k]  = cw - (float)i0;
        iw0[k] = i0;
        iw1[k] = (i0 + 1 >= W_IN) ? 0 : i0 + 1;      // wrap (weight 0 there)
    }

    const bool     issuer = (threadIdx.x < 32);      // wave 0 drives the TDM
    const uint32_t lds0   = (uint32_t)(uintptr_t)(&sbuf[0][0]);
    const uint32_t lds1   = (uint32_t)(uintptr_t)(&sbuf[1][0]);

    // Prologue: kick off tile 0.
    if (issuer) {
        const int ihF0 = (int)floorf((float)rbase * s);
        tdm_issue_rows(psrc, ihF0, lds0);
    }

    for (int t = 0; t < TILES; ++t) {
        const int R       = rbase + t * TILE_ROWS_OUT;
        const int ihFirst = (int)floorf((float)R * s);

#if defined(__HIP_DEVICE_COMPILE__)
        if (issuer) {
            if (t + 1 < TILES) {
                // Overlap: DMA tile t+1 into the other buffer, then wait for
                // tile t (TDM ops complete in order per wave -> cnt<=1 ok).
                const int ihFn = (int)floorf((float)(R + TILE_ROWS_OUT) * s);
                tdm_issue_rows(psrc, ihFn, ((t + 1) & 1) ? lds1 : lds0);
                __builtin_amdgcn_s_wait_tensorcnt(1);
            } else {
                __builtin_amdgcn_s_wait_tensorcnt(0);
            }
        }
#endif
        __syncthreads();          // publish tile t's LDS to all 8 waves

        const float* buf = sbuf[t & 1];
#pragma unroll 4
        for (int rr = 0; rr < TILE_ROWS_OUT; ++rr) {
            const int   row = R + rr;
            const float ch  = (float)row * s;
            const int   ih0 = (int)floorf(ch);
            const float wh  = ch - (float)ih0;       // uniform -> SALU
            const int   li  = ih0 - ihFirst;         // 0..8 within staged tile
            const float* r0 = buf + li * W_IN;
            const float* r1 = r0 + W_IN;
            const size_t obase = ((size_t)plane * H_OUT + (size_t)row) * W_OUT;
#pragma unroll
            for (int k = 0; k < 3; ++k) {
                const float a0 = r0[iw0[k]], a1 = r0[iw1[k]];
                const float b0 = r1[iw0[k]], b1 = r1[iw1[k]];
                const float top = a0 * (1.0f - ww[k]) + a1 * ww[k];
                const float bot = b0 * (1.0f - ww[k]) + b1 * ww[k];
                out[obase + (size_t)((int)threadIdx.x + k * THREADS)] =
                    top * (1.0f - wh) + bot * wh;
            }
        }
        __syncthreads();          // tile t fully consumed before its buffer
                                  // is overwritten by the DMA issued at t+1 (tile t+2)
    }
}

extern "C" void kernel_launch(void* const* d_in, const int* in_sizes, int n_in,
                              void* d_out, int out_size, void* d_ws, size_t ws_size,
                              hipStream_t stream) {
    (void)in_sizes; (void)n_in; (void)out_size; (void)d_ws; (void)ws_size;
    const float* x   = (const float*)d_in[0];
    float*       out = (float*)d_out;

    dim3 grid(NPLANES, SPLIT);   // 128 x 8 = 1024 blocks, 6 pipelined tiles each
    upsample2x_tdm_pipe<<<grid, THREADS, 0, stream>>>(x, out);
}